// GaussianShapedVectorQuantizer_68771016343611
// MI455X (gfx1250) — compile-verified
//
#include <hip/hip_runtime.h>
#include <hip/hip_bf16.h>

typedef __attribute__((ext_vector_type(2))) float v2f;
typedef __attribute__((ext_vector_type(8))) float v8f;

#define NVEC   32768      // N = 32*32*32
#define KCB    1024       // codebook entries
#define DDIM   64         // embedding dim
#define INV_EPS 20.0f     // 1/0.05
#define EPSF   0.05f
#define DUAL_LR 0.5f

// ---------------------------------------------------------------------------
// Kernel 0: zero the histogram
// ---------------------------------------------------------------------------
__global__ void zero_hist(unsigned int* __restrict__ hist) {
    int t = blockIdx.x * 256 + threadIdx.x;
    if (t < KCB) hist[t] = 0u;
}

// ---------------------------------------------------------------------------
// Kernel 1: WMMA f32 distance GEMM + per-row argmin + histogram.
// Block = 256 threads (8 waves); wave owns a 16-row tile of X; block covers
// 128 rows; grid = 256. Codebook k-tiles (16x64 f32) double-buffered in LDS.
// dist(n,k) = ||c_k||^2 - 2 * <x_n, c_k>   (||x_n||^2 constant per row)
// ---------------------------------------------------------------------------
__global__ __launch_bounds__(256) void gemm_argmin(
    const float* __restrict__ x,        // [32,64,32,32]
    const float* __restrict__ cb,       // [1024,64]
    int* __restrict__ idxOut,           // [N]
    unsigned int* __restrict__ hist)    // [K]
{
    __shared__ float Bs[2][16][68];     // pad 4 -> conflict-free b64 reads
    __shared__ float c2s[KCB];

    const int tid  = threadIdx.x;
    const int lane = tid & 31;
    const int wave = tid >> 5;
    const int m    = lane & 15;         // A: row-in-tile; B: column-in-tile
    const int h    = lane >> 4;         // half selects K pair {2h, 2h+1}
    const int rowbase = blockIdx.x * 128 + wave * 16;
    const int n = rowbase + m;

    // ||c_k||^2 per block
    for (int q = 0; q < 4; ++q) {
        int k = tid + q * 256;
        const float* cr = cb + (size_t)k * DDIM;
        float s = 0.f;
        #pragma unroll
        for (int d = 0; d < DDIM; ++d) s += cr[d] * cr[d];
        c2s[k] = s;
    }

    // Preload A (16 rows x 64 depth) per ISA f32 16x4 A layout.
    // x element (n,d) lives at x[(n>>10)*65536 + d*1024 + (n&1023)]
    const float* xcol = x + ((size_t)(n >> 10) * 65536 + (size_t)(n & 1023));
    v2f a[16];
    #pragma unroll
    for (int s = 0; s < 16; ++s) {
        int d = 4 * s + 2 * h;
        a[s].x = xcol[(size_t)d * 1024];
        a[s].y = xcol[(size_t)(d + 1) * 1024];
    }

    // stage k-tile 0
    {
        int e = tid * 4, r = e >> 6, c = e & 63;
        *(float4*)&Bs[0][r][c] = *(const float4*)(cb + (size_t)r * DDIM + c);
    }
    __syncthreads();

    float best[8]; int bestk[8];
    #pragma unroll
    for (int i = 0; i < 8; ++i) { best[i] = 3.4e38f; bestk[i] = 0; }

    for (int kt = 0; kt < 64; ++kt) {
        const int cur = kt & 1;
        float4 nxt;
        if (kt < 63) {
            int e = tid * 4, r = e >> 6, c = e & 63;
            nxt = *(const float4*)(cb + ((size_t)(kt + 1) * 16 + r) * DDIM + c);
            if (kt < 62)
                __builtin_prefetch(cb + ((size_t)(kt + 2) * 16) * DDIM + tid * 4, 0, 0);
        }

        v8f acc = {};
        #pragma unroll
        for (int s = 0; s < 16; ++s) {
            v2f b;
            b.x = Bs[cur][m][4 * s + 2 * h];
            b.y = Bs[cur][m][4 * s + 2 * h + 1];
            acc = __builtin_amdgcn_wmma_f32_16x16x4_f32(
                false, a[s], false, b, (short)0, acc, false, false);
        }

        const int kcol = kt * 16 + m;       // all 8 acc slots share this k
        const float c2v = c2s[kcol];
        #pragma unroll
        for (int i = 0; i < 8; ++i) {
            float dist = c2v - 2.0f * acc[i];
            bestk[i] = (dist < best[i]) ? kcol : bestk[i];
            best[i]  = fminf(dist, best[i]);
        }

        __syncthreads();
        if (kt < 63) {
            int e = tid * 4, r = e >> 6, c = e & 63;
            *(float4*)&Bs[cur ^ 1][r][c] = nxt;
        }
        __syncthreads();
    }

    // Cross-lane argmin within each 16-lane half (C layout: lane = column).
    #pragma unroll
    for (int i = 0; i < 8; ++i) {
        float bv = best[i]; int bk = bestk[i];
        #pragma unroll
        for (int off = 1; off < 16; off <<= 1) {
            float ov = __shfl_xor(bv, off, 32);
            int   ok = __shfl_xor(bk, off, 32);
            bool take = (ov < bv) || (ov == bv && ok < bk);  // lowest-index tie
            bv = take ? ov : bv;
            bk = take ? ok : bk;
        }
        best[i] = bv; bestk[i] = bk;
    }
    if (m == 0) {   // lanes 0 (rows 0..7) and 16 (rows 8..15)
        int rb = rowbase + h * 8;
        #pragma unroll
        for (int i = 0; i < 8; ++i) {
            idxOut[rb + i] = bestk[i];
            atomicAdd(&hist[bestk[i]], 1u);
        }
    }
}

// ---------------------------------------------------------------------------
// Kernel 2: scatter quantized output (codebook gather) + MSE partials.
// out[b,c,hw] = cb[idx[b*1024+hw]][c]; grid = 1024 blocks x 256, 8 elem/thread
// ---------------------------------------------------------------------------
__global__ __launch_bounds__(256) void scatter_mse(
    const float* __restrict__ x, const float* __restrict__ cb,
    const int* __restrict__ idx, float* __restrict__ out,
    float* __restrict__ msep)
{
    __shared__ float red[256];
    float local = 0.f;
    size_t base = (size_t)blockIdx.x * 2048 + threadIdx.x;
    #pragma unroll
    for (int r = 0; r < 8; ++r) {
        size_t o = base + (size_t)r * 256;
        int b  = (int)(o >> 16);
        int c  = (int)(o >> 10) & 63;
        int hw = (int)o & 1023;
        int n  = (b << 10) | hw;
        float q  = cb[(size_t)idx[n] * DDIM + c];
        float xv = x[o];
        out[o] = q;
        float d = q - xv;
        local += d * d;
    }
    red[threadIdx.x] = local; __syncthreads();
    for (int s = 128; s > 0; s >>= 1) {
        if (threadIdx.x < s) red[threadIdx.x] += red[threadIdx.x + s];
        __syncthreads();
    }
    if (threadIdx.x == 0) msep[blockIdx.x] = red[0];
}

// ---------------------------------------------------------------------------
// OT machinery
// ---------------------------------------------------------------------------
__device__ __forceinline__ float blocksum1024(float v, float* red) {
    int t = threadIdx.x;
    red[t] = v; __syncthreads();
    for (int s = 512; s > 0; s >>= 1) {
        if (t < s) red[t] += red[t + s];
        __syncthreads();
    }
    float r = red[0]; __syncthreads();
    return r;
}

__global__ __launch_bounds__(1024) void ot_setup(
    const unsigned int* __restrict__ hist, float* __restrict__ phi,
    float* __restrict__ src, float* __restrict__ tgt,
    float* __restrict__ logtgt)
{
    __shared__ float red[1024];
    int t = threadIdx.x;
    // src = norm_prob(norm_prob(hard_hist))  (double normalize as in reference)
    float hval = (float)hist[t] * (1.0f / (float)NVEC);
    float p  = fmaxf(hval, 1e-12f);
    float S  = blocksum1024(p, red);
    float s1 = p / S;
    float p2 = fmaxf(s1, 1e-12f);
    float S2 = blocksum1024(p2, red);
    float sv = p2 / S2;
    src[t] = sv;
    // gaussian target (mean=(K-1)/2, std=K/6), normalized twice
    float z  = ((float)t - 511.5f) / (1024.0f / 6.0f);
    float tv = expf(-0.5f * z * z);
    float St = blocksum1024(tv, red);
    float t1 = tv / fmaxf(St, 1e-12f);
    float p3 = fmaxf(t1, 1e-12f);
    float S3 = blocksum1024(p3, red);
    float tw = p3 / S3;
    tgt[t] = tw;
    logtgt[t] = logf(fmaxf(tw, 1e-12f));
    phi[t] = 0.f;
}

// per-row logsumexp: lse[i] = LSE_j( logtgt[j] + (phi[j] - |i-j|)/eps )
__global__ __launch_bounds__(256) void ot_lse(
    const float* __restrict__ phi, const float* __restrict__ logtgt,
    float* __restrict__ lse)
{
    __shared__ float red[256];
    int i = blockIdx.x, t = threadIdx.x;
    float a[4];
    #pragma unroll
    for (int q = 0; q < 4; ++q) {
        int j = t + q * 256;
        float cost = fabsf((float)(i - j));
        a[q] = logtgt[j] + (phi[j] - cost) * INV_EPS;
    }
    float m = fmaxf(fmaxf(a[0], a[1]), fmaxf(a[2], a[3]));
    red[t] = m; __syncthreads();
    for (int s = 128; s > 0; s >>= 1) {
        if (t < s) red[t] = fmaxf(red[t], red[t + s]);
        __syncthreads();
    }
    float M = red[0]; __syncthreads();
    float sm = expf(a[0] - M) + expf(a[1] - M) + expf(a[2] - M) + expf(a[3] - M);
    red[t] = sm; __syncthreads();
    for (int s = 128; s > 0; s >>= 1) {
        if (t < s) red[t] += red[t + s];
        __syncthreads();
    }
    if (t == 0) lse[i] = M + logf(red[0]);
}

// dual ascent step: phi[j] += LR * (tgt[j] - sum_i src[i]*exp(a_ij - lse[i]))
__global__ __launch_bounds__(256) void ot_grad(
    float* __restrict__ phi, const float* __restrict__ logtgt,
    const float* __restrict__ src, const float* __restrict__ tgt,
    const float* __restrict__ lse)
{
    __shared__ float red[256];
    int j = blockIdx.x, t = threadIdx.x;
    float phij = phi[j], ltj = logtgt[j];
    float acc = 0.f;
    #pragma unroll
    for (int q = 0; q < 4; ++q) {
        int i = t + q * 256;
        float cost = fabsf((float)(i - j));
        acc += src[i] * expf(ltj + (phij - cost) * INV_EPS - lse[i]);
    }
    red[t] = acc; __syncthreads();
    for (int s = 128; s > 0; s >>= 1) {
        if (t < s) red[t] += red[t + s];
        __syncthreads();
    }
    if (t == 0) phi[j] = phij + DUAL_LR * (tgt[j] - red[0]);
}

// final objective + loss + perplexity
__global__ __launch_bounds__(1024) void finalize_k(
    const unsigned int* __restrict__ hist, const float* __restrict__ src,
    const float* __restrict__ tgt, const float* __restrict__ phi,
    const float* __restrict__ lse, const float* __restrict__ msep,
    float* __restrict__ out_scalars)
{
    __shared__ float red[1024];
    int t = threadIdx.x;
    float s1 = blocksum1024(src[t] * lse[t], red);   // sum src*lse
    float s2 = blocksum1024(tgt[t] * phi[t], red);   // sum tgt*phi
    float avg = (float)hist[t] * (1.0f / (float)NVEC);
    float s3 = blocksum1024(avg * logf(avg + 1e-10f), red);
    float s4 = blocksum1024(msep[t], red);           // total squared error
    if (t == 0) {
        float ot  = -EPSF * s1 + s2;
        float mse = s4 * (1.0f / ((float)NVEC * (float)DDIM));
        out_scalars[0] = mse + 0.25f * mse + 1.0f * ot;  // loss
        out_scalars[1] = expf(-s3);                      // perplexity
    }
}

// ---------------------------------------------------------------------------
extern "C" void kernel_launch(void* const* d_in, const int* in_sizes, int n_in,
                              void* d_out, int out_size, void* d_ws, size_t ws_size,
                              hipStream_t stream) {
    const float* x  = (const float*)d_in[0];   // [32,64,32,32]
    const float* cb = (const float*)d_in[1];   // [1024,64]
    float* out = (float*)d_out;                // 2^21 quantized + loss + perp
    float* ws  = (float*)d_ws;

    int*          idx    = (int*)ws;                      // 32768
    unsigned int* hist   = (unsigned int*)(ws + 32768);   // 1024
    float* msep   = ws + 32768 + 1024;                    // 1024
    float* phi    = msep + 1024;
    float* lse    = phi  + 1024;
    float* src    = lse  + 1024;
    float* tgt    = src  + 1024;
    float* logtgt = tgt  + 1024;

    zero_hist<<<4, 256, 0, stream>>>(hist);
    gemm_argmin<<<256, 256, 0, stream>>>(x, cb, idx, hist);
    scatter_mse<<<1024, 256, 0, stream>>>(x, cb, idx, out, msep);
    ot_setup<<<1, 1024, 0, stream>>>(hist, phi, src, tgt, logtgt);
    for (int s = 0; s < 10; ++s) {
        ot_lse<<<KCB, 256, 0, stream>>>(phi, logtgt, lse);
        ot_grad<<<KCB, 256, 0, stream>>>(phi, logtgt, src, tgt, lse);
    }
    ot_lse<<<KCB, 256, 0, stream>>>(phi, logtgt, lse);
    finalize_k<<<1, 1024, 0, stream>>>(hist, src, tgt, phi, lse, msep,
                                       out + (size_t)32 * 64 * 32 * 32);
}